// FastTensorSRHT_88158498717696
// MI455X (gfx1250) — compile-verified
//
#include <hip/hip_runtime.h>

// FastTensorSRHT for MI455X (gfx1250).
// Memory-bound (~320MB @ 23.3TB/s ~= 14us); FWHT done as H4 (x) H16 (x) H16,
// with the two H16 factors executed on the matrix unit as chained
// V_WMMA_F32_16X16X4_F32 (exact f32 math, +-1 H16 synthesized per-lane from
// popcount -- zero memory traffic for the "weights").
// LDS row stride 20 keeps all 4-float groups 16B-aligned (ds_store_b128) and
// is conflict-free/minimal for both the WMMA B-loads and transposed stores.

#define D_IN    1024
#define D_FEAT  4096
#define DEG     4
#define BATCH   16384
#define RS      20            // padded LDS row stride (16 floats + 4)
#define VSZ     (64 * RS)     // one (row,degree) vector: 64 rows x 16 cols padded
#define ROWS_PER_WG 2

typedef float v2f __attribute__((ext_vector_type(2)));
typedef float v8f __attribute__((ext_vector_type(8)));

__device__ __forceinline__ float hsign(int m, int k) {
  return (__builtin_popcount(m & k) & 1) ? -1.0f : 1.0f;
}

// store a v8f accumulator as two 16B LDS writes (addr must be 16B aligned)
__device__ __forceinline__ void store8(float* p, v8f v) {
  float4 lo = {v[0], v[1], v[2], v[3]};
  float4 hi = {v[4], v[5], v[6], v[7]};
  *(float4*)(p)     = lo;
  *(float4*)(p + 4) = hi;
}

__global__ __launch_bounds__(256) void srht_fwht_wmma(
    const float* __restrict__ x, const float* __restrict__ rad,
    const int* __restrict__ perm, const float* __restrict__ log_ls,
    const float* __restrict__ log_var, float* __restrict__ out)
{
  __shared__ float buf[ROWS_PER_WG * DEG * VSZ];   // 40960 B

  const int tid  = threadIdx.x;
  const int lane = tid & 31;
  const int wave = tid >> 5;          // 8 waves
  const int deg  = wave & 3;
  const int row_local = wave >> 2;    // 0..1
  const int row  = blockIdx.x * ROWS_PER_WG + row_local;

  const float inv_ls = __expf(-log_ls[0]);
  float* vb = &buf[wave * VSZ];

  // ---- Phase 1: stage x*rad*inv_ls into LDS as 64x16 (row r=n>>4, col c=n&15)
  {
    const float4* xv = (const float4*)(x + (size_t)row * D_IN);
    const float4* rv = (const float4*)(rad + deg * D_IN);
    #pragma unroll
    for (int i = 0; i < 8; ++i) {
      int q = lane + 32 * i;               // float4 index, n = 4q
      float4 xx = xv[q];
      float4 rr = rv[q];
      int n = q << 2;
      float4 p;
      p.x = xx.x * rr.x * inv_ls;
      p.y = xx.y * rr.y * inv_ls;
      p.z = xx.z * rr.z * inv_ls;
      p.w = xx.w * rr.w * inv_ls;
      *(float4*)(vb + (n >> 4) * RS + (n & 15)) = p;   // 16B aligned (RS%4==0)
    }
  }

  // ---- H16 as WMMA A-operand chunks (cols k0..k0+3).
  // ISA A layout (32-bit 16x4): VGPR j, lane L -> m = L&15, k = k0 + j + 2*(L>>4)
  const int am = lane & 15;
  const int ak = (lane >> 4) << 1;
  v2f A[4];
  #pragma unroll
  for (int kc = 0; kc < 4; ++kc) {
    A[kc][0] = hsign(am, kc * 4 + ak);
    A[kc][1] = hsign(am, kc * 4 + ak + 1);
  }

  const int bc  = lane & 15;           // B col
  const int bk  = (lane >> 4) << 1;    // B k striping (assumed mirror of A)
  const int dm0 = (lane >> 4) << 3;    // C/D: row M = v + dm0

  // ---- Stage 1: for each slab a (rows a*16..a*16+15): D1 = H16 x T (along b),
  //      store transposed in place -> [c][b'].  LDS is in-order per wave.
  #pragma unroll
  for (int a = 0; a < 4; ++a) {
    float* tb = vb + a * 16 * RS;
    v8f acc = {};
    #pragma unroll
    for (int kc = 0; kc < 4; ++kc) {
      v2f B;
      B[0] = tb[(kc * 4 + bk)     * RS + bc];
      B[1] = tb[(kc * 4 + bk + 1) * RS + bc];
      acc = __builtin_amdgcn_wmma_f32_16x16x4_f32(
          false, A[kc], false, B, (short)0, acc, false, false);
    }
    store8(tb + bc * RS + dm0, acc);     // transpose: [m][c] -> [c][m]
  }

  // ---- Stage 2: D2 = H16 x D1^T (along c); keep all 4 slabs in registers.
  v8f acc2[4];
  #pragma unroll
  for (int a = 0; a < 4; ++a) {
    const float* tb = vb + a * 16 * RS;
    v8f acc = {};
    #pragma unroll
    for (int kc = 0; kc < 4; ++kc) {
      v2f B;
      B[0] = tb[(kc * 4 + bk)     * RS + bc];
      B[1] = tb[(kc * 4 + bk + 1) * RS + bc];
      acc = __builtin_amdgcn_wmma_f32_16x16x4_f32(
          false, A[kc], false, B, (short)0, acc, false, false);
    }
    acc2[a] = acc;
  }

  // ---- Stage 3: H4 along slab index a (pure VALU, elementwise on v8f).
  v8f s01 = acc2[0] + acc2[1];
  v8f d01 = acc2[0] - acc2[1];
  v8f s23 = acc2[2] + acc2[3];
  v8f d23 = acc2[2] - acc2[3];
  v8f f0 = s01 + s23;
  v8f f1 = d01 + d23;
  v8f f2 = s01 - s23;
  v8f f3 = d01 - d23;

  // Register element (lane,v): c' = v + dm0, b' = lane&15.
  // Final vector element n = a'*256 + b'*16 + c' at LDS [(a'*16+b')][c'].
  store8(vb + (0 * 16 + bc) * RS + dm0, f0);
  store8(vb + (1 * 16 + bc) * RS + dm0, f1);
  store8(vb + (2 * 16 + bc) * RS + dm0, f2);
  store8(vb + (3 * 16 + bc) * RS + dm0, f3);

  __syncthreads();

  // ---- Phase 3: permutation gather-product, coalesced float4 output stores.
  // Padded address of element id is (id>>4)*RS + (id&15) == id + 4*(id>>4).
  const float scale = __expf(0.5f * log_var[0]) * 0.015625f;  // 1/sqrt(4096)
  const int rl = tid >> 7;                   // 0..1: output row within WG
  const int t  = tid & 127;
  const float* rb = &buf[rl * DEG * VSZ];
  float4* ov = (float4*)(out + (size_t)(blockIdx.x * ROWS_PER_WG + rl) * D_FEAT);
  #pragma unroll
  for (int it = 0; it < 8; ++it) {
    int f = (t << 2) + it * 512;             // feature base, 4 per thread-iter
    float4 p = {scale, scale, scale, scale};
    #pragma unroll
    for (int d = 0; d < DEG; ++d) {
      int4 id = *(const int4*)(perm + d * D_FEAT + f);
      const float* db = rb + d * VSZ;
      p.x *= db[id.x + ((id.x >> 4) << 2)];
      p.y *= db[id.y + ((id.y >> 4) << 2)];
      p.z *= db[id.z + ((id.z >> 4) << 2)];
      p.w *= db[id.w + ((id.w >> 4) << 2)];
    }
    ov[f >> 2] = p;
  }
}

extern "C" void kernel_launch(void* const* d_in, const int* in_sizes, int n_in,
                              void* d_out, int out_size, void* d_ws, size_t ws_size,
                              hipStream_t stream) {
  (void)in_sizes; (void)n_in; (void)out_size; (void)d_ws; (void)ws_size;
  const float* x    = (const float*)d_in[0];
  const float* rad  = (const float*)d_in[1];
  const int*   perm = (const int*)d_in[2];
  const float* lls  = (const float*)d_in[3];
  const float* lvar = (const float*)d_in[4];
  float* out = (float*)d_out;
  dim3 grid(BATCH / ROWS_PER_WG);   // 8192 workgroups
  dim3 block(256);                  // 8 wave32 waves
  hipLaunchKernelGGL(srht_fwht_wmma, grid, block, 0, stream,
                     x, rad, perm, lls, lvar, out);
}